// PolarBaseClass_18485539242110
// MI455X (gfx1250) — compile-verified
//
#include <hip/hip_runtime.h>

// Problem constants (from reference)
#define B_    4
#define GX_   480
#define GY_   360
#define CIN   64
#define COUT  32
#define NPTS  480000
#define GXY   (GX_ * GY_)          // 172800
#define NSEG  (B_ * GXY)           // 691200

typedef __attribute__((ext_vector_type(2))) float v2f;
typedef __attribute__((ext_vector_type(8))) float v8f;

// Order-preserving float <-> uint key mapping (monotonic: a<b  <=>  key(a)<key(b)).
// key==0 is below every float (even -inf), so it is the scatter-max identity.
__device__ __forceinline__ unsigned f2key(float f) {
    unsigned u = __float_as_uint(f);
    return (u & 0x80000000u) ? ~u : (u | 0x80000000u);
}
__device__ __forceinline__ float key2f(unsigned k) {
    unsigned u = (k & 0x80000000u) ? (k ^ 0x80000000u) : ~k;
    return __uint_as_float(u);   // key 0 -> NaN; masked out by occupancy later
}

// ---------------------------------------------------------------------------
// Pass 1: init workspace. keys (NSEG*CIN uints) = 0, cnt (NSEG uints) = 0.
// 43200 blocks x 256 threads, uint4-vectorized -> exactly 44,236,800 keys.
// ---------------------------------------------------------------------------
__global__ void init_ws(uint4* __restrict__ keys4, unsigned* __restrict__ cnt) {
    int i = blockIdx.x * blockDim.x + threadIdx.x;   // [0, 11059200)
    keys4[i] = uint4{0u, 0u, 0u, 0u};
    if (i < NSEG) cnt[i] = 0u;
}

// ---------------------------------------------------------------------------
// Pass 2: scatter-max. One thread per (point, channel). A wave's 32 lanes hit
// 32 consecutive key addresses of one voxel row -> coalesced u32 max atomics.
// 120000 blocks x 256 threads = exactly NPTS*CIN.
// ---------------------------------------------------------------------------
__global__ void scatter_max(const float* __restrict__ fea,
                            const int*  __restrict__ ind,
                            const int*  __restrict__ bidx,
                            unsigned*   __restrict__ keys,
                            unsigned*   __restrict__ cnt) {
    int gid = blockIdx.x * blockDim.x + threadIdx.x;  // p*64 + ch
    int p  = gid >> 6;
    int ch = gid & 63;
    int seg = bidx[p] * GXY + ind[2 * p] * GY_ + ind[2 * p + 1];
    atomicMax(&keys[(size_t)seg * CIN + ch], f2key(fea[gid]));
    if (ch == 0) atomicAdd(&cnt[seg], 1u);
}

// ---------------------------------------------------------------------------
// Pass 3: fused GEMM(64->32) + bias + ReLU + occupancy mask + NCHW transpose.
// One wave32 per 16-voxel tile; 2 accumulators cover all 32 out channels.
// K=64 as 16 steps of v_wmma_f32_16x16x4_f32 (full f32 precision).
//
// f32 WMMA 16x16x4 operand layout (ISA 7.12.2):
//   A (16x4):  lane m=l&15, half=l>>4 : VGPR0=A[m][2h], VGPR1=A[m][2h+1]
//   B (4x16):  lane n=l&15, half=l>>4 : VGPR0=B[2h][n], VGPR1=B[2h+1][n]
//   C/D:       VGPR r: lanes 0-15 -> (M=r, N=lane), lanes 16-31 -> (M=r+8)
// ---------------------------------------------------------------------------
__global__ __launch_bounds__(128) void gemm_bias_relu_out(
        const unsigned* __restrict__ keys,
        const unsigned* __restrict__ cnt,
        const float*    __restrict__ W,      // [64][32] row-major
        const float*    __restrict__ bias,   // [32]
        float*          __restrict__ out) {  // [B][COUT][GX][GY]
    __shared__ float tile[4][COUT][16];      // per-wave 32ch x 16vox slab (8 KB)

    const int lane = threadIdx.x & 31;
    const int w    = threadIdx.x >> 5;
    const int vbase = (blockIdx.x * 4 + w) * 16;   // first voxel of this tile
    const int m    = lane & 15;
    const int half = lane >> 4;

    // A: pooled row of voxel (vbase+m), starting at k-offset 2*half
    const unsigned* arow = keys + (size_t)(vbase + m) * CIN + 2 * half;
    // B: W[k][n] fragments for n-tiles [0,16) and [16,32)
    const float* b0p = W + (2 * half) * COUT + m;
    const float* b1p = b0p + 16;

    v8f acc0 = {};   // channels  0..15
    v8f acc1 = {};   // channels 16..31
#pragma unroll
    for (int kc = 0; kc < 16; ++kc) {
        const int k0 = kc * 4;
        uint2 av = *(const uint2*)(arow + k0);     // 8B-aligned (offset even)
        v2f a;  a.x = key2f(av.x);      a.y = key2f(av.y);
        v2f b0; b0.x = b0p[k0 * COUT];  b0.y = b0p[(k0 + 1) * COUT];
        v2f b1; b1.x = b1p[k0 * COUT];  b1.y = b1p[(k0 + 1) * COUT];
        acc0 = __builtin_amdgcn_wmma_f32_16x16x4_f32(
                   false, a, false, b0, (short)0, acc0, false, false);
        acc1 = __builtin_amdgcn_wmma_f32_16x16x4_f32(
                   false, a, false, b1, (short)0, acc1, false, false);
    }

    // Stash D tiles into LDS as tile[w][channel][voxel_in_tile]
#pragma unroll
    for (int r = 0; r < 8; ++r) {
        const int mm = r + 8 * half;
        tile[w][m][mm]      = acc0[r];
        tile[w][16 + m][mm] = acc1[r];
    }
    __syncthreads();

    // Coalesced channel-major store: lanes 0-15 -> channel c, 16-31 -> c+1,
    // each covering 16 consecutive y/x positions (64B contiguous per half-wave).
    const unsigned occ = cnt[vbase + m];
    const int posBase  = vbase % GXY;              // tile never crosses a batch
    const size_t outBase = (size_t)(vbase / GXY) * COUT * GXY + posBase + m;
#pragma unroll
    for (int cc = 0; cc < COUT; cc += 2) {
        const int c = cc + half;
        float v = occ ? fmaxf(tile[w][c][m] + bias[c], 0.0f) : 0.0f;
        out[outBase + (size_t)c * GXY] = v;
    }
}

// ---------------------------------------------------------------------------
extern "C" void kernel_launch(void* const* d_in, const int* in_sizes, int n_in,
                              void* d_out, int out_size, void* d_ws, size_t ws_size,
                              hipStream_t stream) {
    const float* pt_fea    = (const float*)d_in[0];   // [N,64] f32
    const int*   pt_ind    = (const int*)  d_in[1];   // [N,2]  i32
    const int*   batch_idx = (const int*)  d_in[2];   // [N]    i32
    const float* W_comp    = (const float*)d_in[3];   // [64,32] f32
    const float* b_comp    = (const float*)d_in[4];   // [32]   f32
    float*       out       = (float*)d_out;           // [4,32,480,360] f32

    // Workspace: 691200*64 u32 keys (176.9 MB) + 691200 u32 counts (2.76 MB)
    unsigned* keys = (unsigned*)d_ws;
    unsigned* cnt  = keys + (size_t)NSEG * CIN;

    init_ws<<<(NSEG * CIN / 4) / 256, 256, 0, stream>>>((uint4*)keys, cnt);
    scatter_max<<<(NPTS * CIN) / 256, 256, 0, stream>>>(pt_fea, pt_ind, batch_idx,
                                                        keys, cnt);
    gemm_bias_relu_out<<<NSEG / 64, 128, 0, stream>>>(keys, cnt, W_comp, b_comp,
                                                      out);
}